// SparseMixerMoeRoutingMethod_10780367913596
// MI455X (gfx1250) — compile-verified
//
#include <hip/hip_runtime.h>
#include <hip/hip_bf16.h>

// SparseMixer top-2 MoE routing for MI455X (gfx1250, wave32).
//
// Mapping: one wave32 per token. 128 experts / 32 lanes = 4 fp32 logits per
// lane, fetched with a single coalesced global_load_b128 per lane (whole
// 512B row per wave). All reductions are in-register wave32 butterflies
// (__shfl_xor); no LDS arrays, no barriers. Loads are non-temporal (pure
// single-pass stream, 128 MiB @ 23.3 TB/s ~ 5.7 us floor). Each wave
// processes two tokens and prefetches the second row (global_prefetch_b8)
// before crunching the first.

#define NEXP   128
#define TOPK   2
#define EPSV   0.01f
#define WPB    8                 // waves per block
#define BLOCKSZ (32 * WPB)       // 256 threads = 8 wave32
#define TOK_PER_BLOCK (2 * WPB)  // 2 tokens per wave

typedef float v4f __attribute__((ext_vector_type(4)));

__global__ __launch_bounds__(BLOCKSZ) void sparsemixer_top2_kernel(
    const float* __restrict__ logits,
    int* __restrict__ idx_out,
    float* __restrict__ val_out,
    int n_tokens)
{
    const int lane = threadIdx.x & 31;
    const int wave = threadIdx.x >> 5;
    const int base = blockIdx.x * TOK_PER_BLOCK;

    // Prefetch this wave's SECOND row while we process the first.
    {
        const int t1 = base + WPB + wave;
        if (t1 < n_tokens)
            __builtin_prefetch(logits + (size_t)t1 * NEXP + lane * 4, 0, 0);
    }

#pragma unroll
    for (int half = 0; half < 2; ++half) {
        const int token = base + half * WPB + wave;
        if (token < n_tokens) {
            const v4f* row4 = (const v4f*)(logits + (size_t)token * NEXP);
            // Non-temporal: data is streamed exactly once.
            v4f v = __builtin_nontemporal_load(row4 + lane);
            float l[4] = {v.x, v.y, v.z, v.w};

            int   ridx[TOPK];
            float rval[TOPK];

#pragma unroll
            for (int k = 0; k < TOPK; ++k) {
                // ---- argmax (value, lowest index on tie) ----
                float bv = l[0];
                int   bi = lane * 4;
#pragma unroll
                for (int j = 1; j < 4; ++j) {
                    if (l[j] > bv) { bv = l[j]; bi = lane * 4 + j; }
                }
#pragma unroll
                for (int off = 16; off; off >>= 1) {
                    float ov = __shfl_xor(bv, off, 32);
                    int   oi = __shfl_xor(bi, off, 32);
                    if (ov > bv || (ov == bv && oi < bi)) { bv = ov; bi = oi; }
                }

                // ---- masked softmax value at the argmax ----
                // sm[max_idx] = 1 / sum_{j unmasked} exp(l_j - max).
                // Knocked-out -inf entries: mask expr is NaN -> "unmasked",
                // but exp(-inf)=0, identical to the reference semantics.
                float s = 0.0f;
#pragma unroll
                for (int j = 0; j < 4; ++j) {
                    const float diff = bv - l[j];
                    const float den  = fmaxf(fabsf(l[j]), bv);
                    const bool masked = (diff / den) > (2.0f * EPSV);
                    s += masked ? 0.0f : expf(l[j] - bv);
                }
#pragma unroll
                for (int off = 16; off; off >>= 1)
                    s += __shfl_xor(s, off, 32);

                ridx[k] = bi;
                rval[k] = 1.0f / s;

                // Knock out the winner for the next round (lane that owns it).
                if ((bi >> 2) == lane) l[bi & 3] = -__builtin_inff();
            }

            if (lane == 0) {
                *(int2*)(idx_out + (size_t)token * TOPK)   = make_int2(ridx[0], ridx[1]);
                *(float2*)(val_out + (size_t)token * TOPK) = make_float2(rval[0], rval[1]);
            }
        }
    }
}

extern "C" void kernel_launch(void* const* d_in, const int* in_sizes, int n_in,
                              void* d_out, int out_size, void* d_ws, size_t ws_size,
                              hipStream_t stream)
{
    (void)n_in; (void)out_size; (void)d_ws; (void)ws_size;

    const float* logits = (const float*)d_in[0];
    const int n_tokens  = in_sizes[0] / NEXP;

    // d_out = [topk_indices (T*2 int32) | topk_values (T*2 f32)], flat.
    int*   idx_out = (int*)d_out;
    float* val_out = (float*)d_out + (size_t)n_tokens * TOPK;

    const int grid = (n_tokens + TOK_PER_BLOCK - 1) / TOK_PER_BLOCK;
    hipLaunchKernelGGL(sparsemixer_top2_kernel, dim3(grid), dim3(BLOCKSZ), 0, stream,
                       logits, idx_out, val_out, n_tokens);
}